// ContrastiveLoss_77738908057913
// MI455X (gfx1250) — compile-verified
//
#include <hip/hip_runtime.h>
#include <hip/hip_bf16.h>
#include <math.h>

// ---------------------------------------------------------------------------
// Fused contrastive loss for MI455X (gfx1250, wave32).
// Bandwidth-bound: one streaming pass reads each input element exactly once,
// computing row reductions + the 32x32 Gram matrix via V_WMMA_F32_16X16X4_F32.
// Sigmoid uses v_exp_f32 + v_rcp_f32 (no IEEE div expansion) to keep the
// inner loop VALU-light and memory-bound.
// ---------------------------------------------------------------------------

typedef __attribute__((ext_vector_type(2))) float v2f;
typedef __attribute__((ext_vector_type(8))) float v8f;

#define TAU_F     0.1f
#define PART_LEN  1120    // 32*32 cross + 32 sq1 + 32 sq2 + 32 masked-diff
#define G2        32      // workgroups in reduction pass 2

__device__ __forceinline__ float sigmoidf1(float x) {
    // 1/(1+e^-x) with fast hardware reciprocal: v_mul + v_exp_f32 + v_add +
    // v_rcp_f32 instead of the ~10-op IEEE divide sequence.
    return __builtin_amdgcn_rcpf(1.0f + __expf(-x));
}

__device__ __forceinline__ v2f sig2(v2f v) {
    v2f r;
    r.x = sigmoidf1(v.x);
    r.y = sigmoidf1(v.y);
    return r;
}

// Load one 16x4 f32 WMMA fragment (A layout; B=f2^T uses the identical
// per-lane pattern): lane l<16 -> row l, k pair (k0,k0+1); lane l+16 -> row l,
// k pair (k0+2,k0+3). k0 is a multiple of 4 so the b64 load is 8B-aligned.
__device__ __forceinline__ v2f load_frag(const float* __restrict__ p,
                                         int rowBase, int k0, int N, int lane) {
    int row  = rowBase + (lane & 15);
    int koff = k0 + ((lane >> 4) << 1);
    return *reinterpret_cast<const v2f*>(p + (size_t)row * (size_t)N + koff);
}

// -------------------------- Pass 1: streaming ------------------------------
__global__ __launch_bounds__(256) void cl_pass1(const float* __restrict__ in1,
                                                const float* __restrict__ in2,
                                                const float* __restrict__ msk,
                                                float* __restrict__ part,
                                                int N, int kPerWG) {
    __shared__ float s_cross[32][32];
    __shared__ float s_sq1[32];
    __shared__ float s_sq2[32];
    __shared__ float s_md[32];

    const int tid = threadIdx.x;
    for (int i = tid; i < 1024; i += 256) (&s_cross[0][0])[i] = 0.0f;
    if (tid < 32) { s_sq1[tid] = 0.0f; s_sq2[tid] = 0.0f; s_md[tid] = 0.0f; }
    __syncthreads();

    const int lane  = tid & 31;
    const int wave  = tid >> 5;           // 0..7
    const int kWave = kPerWG >> 3;        // per-wave K chunk (multiple of 4)
    int       k0    = blockIdx.x * kPerWG + wave * kWave;
    const int kEnd  = k0 + kWave;

    v8f c00 = {}; v8f c01 = {}; v8f c10 = {}; v8f c11 = {};
    float s1lo = 0.f, s1hi = 0.f, s2lo = 0.f, s2hi = 0.f, dlo = 0.f, dhi = 0.f;

    for (; k0 < kEnd; k0 += 4) {
        v2f a0 = sig2(load_frag(in1,  0, k0, N, lane));  // x1 rows 0..15
        v2f a1 = sig2(load_frag(in1, 16, k0, N, lane));  // x1 rows 16..31
        v2f b0 = sig2(load_frag(in2,  0, k0, N, lane));  // x2 rows 0..15
        v2f b1 = sig2(load_frag(in2, 16, k0, N, lane));  // x2 rows 16..31
        v2f m0 =      load_frag(msk,  0, k0, N, lane);
        v2f m1 =      load_frag(msk, 16, k0, N, lane);

        // per-row reductions (each lane covers 2 k's of one row)
        s1lo += a0.x * a0.x + a0.y * a0.y;
        s1hi += a1.x * a1.x + a1.y * a1.y;
        s2lo += b0.x * b0.x + b0.y * b0.y;
        s2hi += b1.x * b1.x + b1.y * b1.y;
        float t0 = m0.x * (a0.x - b0.x), t1 = m0.y * (a0.y - b0.y);
        dlo += t0 * t0 + t1 * t1;
        t0 = m1.x * (a1.x - b1.x); t1 = m1.y * (a1.y - b1.y);
        dhi += t0 * t0 + t1 * t1;

        // cross[i][j] += sum_k x1[i,k] * x2[j,k] : four 16x16 tiles, K=4/step
        c00 = __builtin_amdgcn_wmma_f32_16x16x4_f32(false, a0, false, b0,
                                                    (short)0, c00, false, false);
        c01 = __builtin_amdgcn_wmma_f32_16x16x4_f32(false, a0, false, b1,
                                                    (short)0, c01, false, false);
        c10 = __builtin_amdgcn_wmma_f32_16x16x4_f32(false, a1, false, b0,
                                                    (short)0, c10, false, false);
        c11 = __builtin_amdgcn_wmma_f32_16x16x4_f32(false, a1, false, b1,
                                                    (short)0, c11, false, false);
    }

    // fold lane l <-> l+16 (both halves accumulated the same row)
    s1lo += __shfl_xor(s1lo, 16, 32);
    s1hi += __shfl_xor(s1hi, 16, 32);
    s2lo += __shfl_xor(s2lo, 16, 32);
    s2hi += __shfl_xor(s2hi, 16, 32);
    dlo  += __shfl_xor(dlo,  16, 32);
    dhi  += __shfl_xor(dhi,  16, 32);
    if (lane < 16) {
        atomicAdd(&s_sq1[lane],      s1lo);
        atomicAdd(&s_sq1[lane + 16], s1hi);
        atomicAdd(&s_sq2[lane],      s2lo);
        atomicAdd(&s_sq2[lane + 16], s2hi);
        atomicAdd(&s_md[lane],       dlo);
        atomicAdd(&s_md[lane + 16],  dhi);
    }

    // accumulate C tiles: VGPR v -> M = v (lanes 0..15) / v+8 (lanes 16..31),
    // N = lane & 15
    {
        int n  = lane & 15;
        int mB = (lane < 16) ? 0 : 8;
#pragma unroll
        for (int v = 0; v < 8; ++v) {
            atomicAdd(&s_cross[mB + v][n],            c00[v]);
            atomicAdd(&s_cross[mB + v][16 + n],       c01[v]);
            atomicAdd(&s_cross[16 + mB + v][n],       c10[v]);
            atomicAdd(&s_cross[16 + mB + v][16 + n],  c11[v]);
        }
    }
    __syncthreads();

    float* dst = part + (size_t)blockIdx.x * PART_LEN;
    for (int i = tid; i < 1024; i += 256) dst[i] = (&s_cross[0][0])[i];
    if (tid < 32) {
        dst[1024 + tid] = s_sq1[tid];
        dst[1056 + tid] = s_sq2[tid];
        dst[1088 + tid] = s_md[tid];
    }
}

// ---------------------- Pass 2: tree reduce partials -----------------------
__global__ __launch_bounds__(256) void cl_pass2(const float* __restrict__ part,
                                                float* __restrict__ part2,
                                                int wgPerBlock) {
    const float* src = part + (size_t)blockIdx.x * wgPerBlock * PART_LEN;
    float*       dst = part2 + (size_t)blockIdx.x * PART_LEN;
    for (int e = threadIdx.x; e < PART_LEN; e += 256) {
        float s = 0.0f;
        for (int w = 0; w < wgPerBlock; ++w) s += src[(size_t)w * PART_LEN + e];
        dst[e] = s;
    }
}

// ---------------------- Pass 3: final loss evaluation ----------------------
__global__ __launch_bounds__(128) void cl_pass3(const float* __restrict__ part2,
                                                float* __restrict__ out, int N) {
    __shared__ float acc[PART_LEN];
    __shared__ float lossArr[32];

    for (int e = threadIdx.x; e < PART_LEN; e += 128) {
        float s = 0.0f;
        for (int b = 0; b < G2; ++b) s += part2[(size_t)b * PART_LEN + e];
        acc[e] = s;
    }
    __syncthreads();

    if (threadIdx.x < 32) {
        const int   i    = threadIdx.x;
        const float invN = 1.0f / (float)N;
        const float sq1  = acc[1024 + i];
        float simsum = 0.0f, diag = 0.0f;
        for (int j = 0; j < 32; ++j) {
            float d = (sq1 + acc[1056 + j] - 2.0f * acc[i * 32 + j]) * invN;
            float s = __expf(-d * (1.0f / TAU_F));
            simsum += s;
            if (j == i) diag = s;
        }
        float sim_neg = simsum - diag;
        float sim_pos = __expf(-(acc[1088 + i] * invN) * (1.0f / TAU_F));
        lossArr[i] = -__logf(sim_pos * __builtin_amdgcn_rcpf(sim_pos + sim_neg));
    }
    __syncthreads();

    if (threadIdx.x == 0) {
        float s = 0.0f;
        for (int i = 0; i < 32; ++i) s += lossArr[i];
        out[0] = s * (1.0f / 32.0f);
    }
}

// ---------------------------------------------------------------------------
extern "C" void kernel_launch(void* const* d_in, const int* in_sizes, int n_in,
                              void* d_out, int out_size, void* d_ws, size_t ws_size,
                              hipStream_t stream) {
    const float* in1 = (const float*)d_in[0];
    const float* in2 = (const float*)d_in[1];
    const float* msk = (const float*)d_in[2];
    float*       out = (float*)d_out;

    const int B    = 32;
    const int NTOT = in_sizes[0];
    const int N    = NTOT / B;          // 262144

    // Choose pass-1 grid: largest G fitting workspace; N divides by all.
    int G = 512;
    while (G > G2 && ((size_t)(G + G2) * PART_LEN * sizeof(float)) > ws_size)
        G >>= 1;
    const int kPerWG = N / G;           // 512 at G=512; multiple of 32

    float* part  = (float*)d_ws;
    float* part2 = part + (size_t)G * PART_LEN;

    cl_pass1<<<G, 256, 0, stream>>>(in1, in2, msk, part, N, kPerWG);
    cl_pass2<<<G2, 256, 0, stream>>>(part, part2, G / G2);
    cl_pass3<<<1, 128, 0, stream>>>(part2, out, N);
}